// PoleInteraction_62440234549688
// MI455X (gfx1250) — compile-verified
//
#include <hip/hip_runtime.h>
#include <hip/hip_bf16.h>

#define NCH 64   // channels

typedef float v2f __attribute__((ext_vector_type(2)));
typedef float v8f __attribute__((ext_vector_type(8)));

// ---------------------------------------------------------------------------
// Kernel 0: zero the output accumulator (segment_sum target).
// ---------------------------------------------------------------------------
__global__ __launch_bounds__(256) void zero_out_kernel(float4* __restrict__ out, int n4) {
    int i = blockIdx.x * blockDim.x + threadIdx.x;
    if (i < n4) out[i] = make_float4(0.f, 0.f, 0.f, 0.f);
}

// ---------------------------------------------------------------------------
// Kernel 1: per-node LayerNorm statistics. One wave32 per node; each lane
// loads 2 of the 64 channels, butterfly-reduce sum/sumsq across the wave.
// ---------------------------------------------------------------------------
__global__ __launch_bounds__(256) void ln_stats_kernel(const float* __restrict__ x,
                                                       float* __restrict__ mean,
                                                       float* __restrict__ rstd,
                                                       int n) {
    int wave = (blockIdx.x * blockDim.x + threadIdx.x) >> 5;
    int lane = threadIdx.x & 31;
    if (wave >= n) return;
    const float* xr = x + (size_t)wave * NCH;
    float a = xr[lane];
    float b = xr[lane + 32];
    float s  = a + b;
    float sq = a * a + b * b;
    #pragma unroll
    for (int off = 16; off >= 1; off >>= 1) {
        s  += __shfl_xor(s,  off, 32);
        sq += __shfl_xor(sq, off, 32);
    }
    if (lane == 0) {
        float m = s * (1.0f / NCH);
        float v = sq * (1.0f / NCH) - m * m;          // population variance (jnp.var)
        mean[wave] = m;
        rstd[wave] = rsqrtf(v + 1e-5f);
    }
}

// ---------------------------------------------------------------------------
// Kernel 2: fused LayerNorm-apply + MLP with V_WMMA_F32_16X16X4_F32.
// One wave32 owns 16 node rows. GEMM1 (K=64, 8 col tiles of the 128 hidden)
// keeps the LN'd A fragments in registers; silu'd hidden tile goes to a
// wave-private 16x128 LDS slab; GEMM2 (K=128, 4 col tiles) reads A from LDS.
// Output q = (h@w2 + b2) * vln_weight.
//
// WMMA f32 layouts (ISA 7.12.2, wave32):
//   A (16x4):  lane l -> row (l&15), vgpr v -> K = 2*(l>>4) + v
//   B (4x16):  lane l -> col (l&15), vgpr v -> K = 2*(l>>4) + v
//   C/D (16x16): lane l, vgpr v -> row v + 8*(l>>4), col (l&15)
// ---------------------------------------------------------------------------
__global__ __launch_bounds__(128) void mlp_wmma_kernel(
    const float* __restrict__ x,
    const float* __restrict__ mean, const float* __restrict__ rstd,
    const float* __restrict__ ln_scale, const float* __restrict__ ln_bias,
    const float* __restrict__ w1, const float* __restrict__ b1,
    const float* __restrict__ w2, const float* __restrict__ b2,
    const float* __restrict__ vln,
    float* __restrict__ q, int n) {
    __shared__ float hsh[4][16 * 128];   // 32 KB, one 16x128 f32 tile per wave

    const int wave = threadIdx.x >> 5;
    const int lane = threadIdx.x & 31;
    const int rowBase = blockIdx.x * 64 + wave * 16;
    if (rowBase >= n) return;            // uniform per wave -> EXEC all-ones inside

    const int mloc  = lane & 15;         // row within tile (A) or col (B/C)
    const int khalf = (lane >> 4) * 2;   // K offset from lane half
    const int hrow  = 8 * (lane >> 4);   // C/D row offset from lane half
    const int mrow  = rowBase + mloc;

    const float mu = mean[mrow];
    const float rs = rstd[mrow];

    // --- A fragments for GEMM1 with LayerNorm fused (kept in VGPRs) ---
    v2f afrag[16];
    #pragma unroll
    for (int kk = 0; kk < 16; ++kk) {
        int k0 = kk * 4 + khalf;
        float x0 = x[(size_t)mrow * NCH + k0];
        float x1 = x[(size_t)mrow * NCH + k0 + 1];
        afrag[kk].x = (x0 - mu) * rs * ln_scale[k0]     + ln_bias[k0];
        afrag[kk].y = (x1 - mu) * rs * ln_scale[k0 + 1] + ln_bias[k0 + 1];
    }

    // --- GEMM1: h = silu(x_norm @ w1 + b1), tile-by-tile into LDS ---
    #pragma unroll
    for (int ct = 0; ct < 8; ++ct) {
        const int ncol = ct * 16 + mloc;           // hidden column 0..127
        v8f acc = {0.f, 0.f, 0.f, 0.f, 0.f, 0.f, 0.f, 0.f};
        #pragma unroll
        for (int kk = 0; kk < 16; ++kk) {
            int kb = kk * 4 + khalf;
            v2f bfrag;
            bfrag.x = w1[(size_t)kb       * 128 + ncol];
            bfrag.y = w1[(size_t)(kb + 1) * 128 + ncol];
            acc = __builtin_amdgcn_wmma_f32_16x16x4_f32(
                false, afrag[kk], false, bfrag, (short)0, acc, false, false);
        }
        const float bias = b1[ncol];
        #pragma unroll
        for (int v = 0; v < 8; ++v) {
            float t = acc[v] + bias;
            float h = t / (1.0f + __expf(-t));     // silu
            hsh[wave][(v + hrow) * 128 + ncol] = h;
        }
    }

    // --- GEMM2: q = (h @ w2 + b2) * vln_weight ---
    #pragma unroll
    for (int ct = 0; ct < 4; ++ct) {
        const int ncol = ct * 16 + mloc;           // output column 0..63
        v8f acc = {0.f, 0.f, 0.f, 0.f, 0.f, 0.f, 0.f, 0.f};
        #pragma unroll
        for (int kk = 0; kk < 32; ++kk) {
            int kb = kk * 4 + khalf;
            v2f a2, b2f;
            a2.x  = hsh[wave][mloc * 128 + kb];
            a2.y  = hsh[wave][mloc * 128 + kb + 1];
            b2f.x = w2[(size_t)kb       * NCH + ncol];
            b2f.y = w2[(size_t)(kb + 1) * NCH + ncol];
            acc = __builtin_amdgcn_wmma_f32_16x16x4_f32(
                false, a2, false, b2f, (short)0, acc, false, false);
        }
        const float bias = b2[ncol];
        const float wv   = vln[ncol];
        #pragma unroll
        for (int v = 0; v < 8; ++v) {
            int row = rowBase + v + hrow;
            q[(size_t)row * NCH + ncol] = (acc[v] + bias) * wv;
        }
    }
}

// ---------------------------------------------------------------------------
// Kernel 3: edge interaction + scatter-add. One wave32 per edge, 2 channels
// per lane (float2). vln_weight already folded into q, so:
//   dE_c = q_i[c] * q_j[c] * sum_d vec_i[d][c] * (M @ vec_j)[d][c]
// Scatter with native f32 global atomics (data fits in 192MB L2).
// ---------------------------------------------------------------------------
__global__ __launch_bounds__(256) void edge_kernel(
    const int* __restrict__ senders, const int* __restrict__ receivers,
    const float* __restrict__ Mmat, const float* __restrict__ vec,
    const float* __restrict__ q, float* __restrict__ out, int ne) {
    int e    = (blockIdx.x * blockDim.x + threadIdx.x) >> 5;
    int lane = threadIdx.x & 31;
    if (e >= ne) return;

    const int s = senders[e];
    const int r = receivers[e];

    const float* Me = Mmat + (size_t)e * 9;
    float m00 = Me[0], m01 = Me[1], m02 = Me[2];
    float m10 = Me[3], m11 = Me[4], m12 = Me[5];
    float m20 = Me[6], m21 = Me[7], m22 = Me[8];

    const int c = lane * 2;
    const float* vi = vec + (size_t)r * 3 * NCH + c;
    const float* vj = vec + (size_t)s * 3 * NCH + c;

    float2 qi  = *(const float2*)(q + (size_t)r * NCH + c);
    float2 qj  = *(const float2*)(q + (size_t)s * NCH + c);
    float2 vi0 = *(const float2*)(vi + 0 * NCH);
    float2 vi1 = *(const float2*)(vi + 1 * NCH);
    float2 vi2 = *(const float2*)(vi + 2 * NCH);
    float2 vj0 = *(const float2*)(vj + 0 * NCH);
    float2 vj1 = *(const float2*)(vj + 1 * NCH);
    float2 vj2 = *(const float2*)(vj + 2 * NCH);

    // field = M @ vec_j (per channel), q_j scalar factor folded at the end
    float f0x = m00 * vj0.x + m01 * vj1.x + m02 * vj2.x;
    float f1x = m10 * vj0.x + m11 * vj1.x + m12 * vj2.x;
    float f2x = m20 * vj0.x + m21 * vj1.x + m22 * vj2.x;
    float f0y = m00 * vj0.y + m01 * vj1.y + m02 * vj2.y;
    float f1y = m10 * vj0.y + m11 * vj1.y + m12 * vj2.y;
    float f2y = m20 * vj0.y + m21 * vj1.y + m22 * vj2.y;

    float ex = qi.x * qj.x * (vi0.x * f0x + vi1.x * f1x + vi2.x * f2x);
    float ey = qi.y * qj.y * (vi0.y * f0y + vi1.y * f1y + vi2.y * f2y);

    float* o = out + (size_t)r * NCH + c;
    unsafeAtomicAdd(o,     ex);
    unsafeAtomicAdd(o + 1, ey);
}

// ---------------------------------------------------------------------------
// Host-side launcher
// ---------------------------------------------------------------------------
extern "C" void kernel_launch(void* const* d_in, const int* in_sizes, int n_in,
                              void* d_out, int out_size, void* d_ws, size_t ws_size,
                              hipStream_t stream) {
    const float* x        = (const float*)d_in[0];
    const float* vec      = (const float*)d_in[1];
    const int*   senders  = (const int*)d_in[2];
    const int*   receivers= (const int*)d_in[3];
    const float* inter    = (const float*)d_in[4];
    const float* ln_scale = (const float*)d_in[5];
    const float* ln_bias  = (const float*)d_in[6];
    const float* vln      = (const float*)d_in[7];
    const float* w1       = (const float*)d_in[8];
    const float* b1       = (const float*)d_in[9];
    const float* w2       = (const float*)d_in[10];
    const float* b2       = (const float*)d_in[11];
    float* out = (float*)d_out;

    const int n  = in_sizes[0] / NCH;   // nodes
    const int ne = in_sizes[2];         // edges

    // workspace layout: mean[n] | rstd[n] | q[n*NCH]   (~13.2 MB for n=50000)
    float* mean = (float*)d_ws;
    float* rstd = mean + n;
    float* q    = rstd + n;

    const int n4 = (n * NCH) / 4;
    zero_out_kernel<<<(n4 + 255) / 256, 256, 0, stream>>>((float4*)out, n4);
    ln_stats_kernel<<<(n + 7) / 8, 256, 0, stream>>>(x, mean, rstd, n);
    mlp_wmma_kernel<<<(n + 63) / 64, 128, 0, stream>>>(
        x, mean, rstd, ln_scale, ln_bias, w1, b1, w2, b2, vln, q, n);
    edge_kernel<<<(ne + 7) / 8, 256, 0, stream>>>(senders, receivers, inter, vec, q, out, ne);
}